// MultiPositiveContrastiveLoss_21380347200380
// MI455X (gfx1250) — compile-verified
//
#include <hip/hip_runtime.h>

// MultiPositiveContrastiveLoss for MI455X (gfx1250, wave32).
// Pair matrix (margin - pos_i + neg_j) is rank-2 => built with
// V_WMMA_F32_16X16X4_F32:  A = [c | 1 | 0 | 0] (16x4), B = [1s ; n ; 0 ; 0] (4x16).
// relu via single v_med3_f32 (finite clamp so InstCombine can't fold to maxnum);
// accumulate via v_pk_add_f32 (co-executes with the matrix pipe).
// The problem loop index is forced scalar (readfirstlane) so counts load via
// SMEM, bases live in SGPRs (saddr-form gathers), branches are SALU, and EXEC
// is provably all-1s at every WMMA (ISA requirement).

typedef __attribute__((ext_vector_type(2))) float v2f;
typedef __attribute__((ext_vector_type(8))) float v8f;

#define MARGIN_F 0.5f
#define BPROB 32768
#define CAND 64
#define MAXP 16
#define MAXN 48
#define BLOCKS 512
#define THREADS 256
#define WAVES_PER_BLOCK (THREADS / 32)
#define NEG_SENTINEL (-1.0e30f)

__device__ __forceinline__ float relu1(float x) {
    // clamp(x, 0, 3e38): single v_med3_f32; finite hi bound avoids the
    // fmed3->fmaxnum InstCombine fold (which costs an extra canonicalize max).
    return __builtin_amdgcn_fmed3f(x, 0.0f, 3.0e38f);
}

__global__ __launch_bounds__(THREADS) void mpcl_pair_kernel(
    const float* __restrict__ scores,
    const int* __restrict__ pos_indices,
    const int* __restrict__ neg_indices,
    const int* __restrict__ pos_counts,
    const int* __restrict__ neg_counts,
    float* __restrict__ partial_sum,
    unsigned int* __restrict__ partial_cnt)
{
    const int lane   = threadIdx.x & 31;
    const int wib    = threadIdx.x >> 5;
    // Force the per-wave problem index scalar: SGPR loop, SMEM count loads,
    // SGPR base addresses for every gather.
    const int gwave  = __builtin_amdgcn_readfirstlane(
                           blockIdx.x * WAVES_PER_BLOCK + wib);
    const int nwaves = gridDim.x * WAVES_PER_BLOCK;
    const int lanelo = lane & 15;
    const bool lohalf = (lane < 16);

    v8f acc = {};
    unsigned int cnt = 0;
    const float onesRow = lohalf ? 1.0f : 0.0f;  // B row 0 (K=0) / zeros for K=2
    const v8f cz = {};

    for (int b = gwave; b < BPROB; b += nwaves) {  // scalar loop
        const int pc = pos_counts[b];              // s_load_b32
        const int nc = neg_counts[b];              // s_load_b32

        // Prefetch next iteration's lines while we compute this one.
        const int bn = b + nwaves;
        if (bn < BPROB) {
            __builtin_prefetch(&scores[bn * CAND], 0, 0);
            __builtin_prefetch(&neg_indices[bn * MAXN], 0, 0);
            __builtin_prefetch(&pos_indices[bn * MAXP], 0, 0);
        }

        if (pc == 0 || nc == 0) continue;          // scalar branch
        cnt += (unsigned int)(pc * nc);

        const float* sb = scores + b * CAND;       // SGPR base

        // ---- A matrix: lanes 0-15 hold {margin - pos, 1}; lanes 16-31 {0,0}
        const int pidx = pos_indices[b * MAXP + lanelo];
        const float p  = sb[pidx];
        const float cI = (lanelo < pc) ? (MARGIN_F - p) : NEG_SENTINEL;
        v2f a;
        a.x = lohalf ? cI : 0.0f;
        a.y = onesRow;

        // ---- negatives: lanes 0-31 hold n[0..31] (masked with sentinel)
        const int nidxA = neg_indices[b * MAXN + lane];
        float nA = sb[nidxA];
        nA = (lane < nc) ? nA : NEG_SENTINEL;

        v2f bm;
        bm.x = onesRow;

        // tile 0: negs 0..15
        bm.y = lohalf ? nA : 0.0f;
        v8f d = __builtin_amdgcn_wmma_f32_16x16x4_f32(
            false, a, false, bm, (short)0, cz, false, false);
#pragma unroll
        for (int k = 0; k < 8; ++k) acc[k] += relu1(d[k]);

        // tile 1: negs 16..31 (scalar skip when nc <= 16)
        if (nc > 16) {
            const float nMid = __shfl(nA, lanelo + 16, 32);
            bm.y = lohalf ? nMid : 0.0f;
            d = __builtin_amdgcn_wmma_f32_16x16x4_f32(
                false, a, false, bm, (short)0, cz, false, false);
#pragma unroll
            for (int k = 0; k < 8; ++k) acc[k] += relu1(d[k]);
        }

        // tile 2: negs 32..47 (scalar skip when nc <= 32)
        if (nc > 32) {
            const int nidxB = neg_indices[b * MAXN + 32 + lanelo];
            float nB = sb[nidxB];
            nB = ((32 + lanelo) < nc) ? nB : NEG_SENTINEL;
            bm.y = lohalf ? nB : 0.0f;
            d = __builtin_amdgcn_wmma_f32_16x16x4_f32(
                false, a, false, bm, (short)0, cz, false, false);
#pragma unroll
            for (int k = 0; k < 8; ++k) acc[k] += relu1(d[k]);
        }
    }

    // ---- wave reduction (fixed order)
    float s = 0.0f;
#pragma unroll
    for (int k = 0; k < 8; ++k) s += acc[k];
#pragma unroll
    for (int m = 16; m >= 1; m >>= 1) s += __shfl_xor(s, m, 32);

    __shared__ float        wsum[WAVES_PER_BLOCK];
    __shared__ unsigned int wcnt[WAVES_PER_BLOCK];
    if (lane == 0) { wsum[wib] = s; wcnt[wib] = cnt; }
    __syncthreads();
    if (threadIdx.x == 0) {
        float t = 0.0f;
        unsigned int c = 0;
        for (int i = 0; i < WAVES_PER_BLOCK; ++i) { t += wsum[i]; c += wcnt[i]; }
        partial_sum[blockIdx.x] = t;
        partial_cnt[blockIdx.x] = c;
    }
}

__global__ __launch_bounds__(256) void mpcl_reduce_kernel(
    const float* __restrict__ partial_sum,
    const unsigned int* __restrict__ partial_cnt,
    float* __restrict__ out)
{
    __shared__ float        ssum[256];
    __shared__ unsigned int scnt[256];
    const int t = threadIdx.x;
    ssum[t] = partial_sum[t] + partial_sum[t + 256];   // BLOCKS == 512
    scnt[t] = partial_cnt[t] + partial_cnt[t + 256];
    __syncthreads();
    for (int step = 128; step >= 1; step >>= 1) {
        if (t < step) { ssum[t] += ssum[t + step]; scnt[t] += scnt[t + step]; }
        __syncthreads();
    }
    if (t == 0) {
        out[0] = (scnt[0] > 0) ? (ssum[0] / (float)scnt[0]) : 0.0f;
    }
}

extern "C" void kernel_launch(void* const* d_in, const int* in_sizes, int n_in,
                              void* d_out, int out_size, void* d_ws, size_t ws_size,
                              hipStream_t stream) {
    const float* scores      = (const float*)d_in[0];
    const int*   pos_indices = (const int*)d_in[1];
    const int*   neg_indices = (const int*)d_in[2];
    const int*   pos_counts  = (const int*)d_in[3];
    const int*   neg_counts  = (const int*)d_in[4];

    float*        partial_sum = (float*)d_ws;
    unsigned int* partial_cnt = (unsigned int*)((char*)d_ws + BLOCKS * sizeof(float));

    mpcl_pair_kernel<<<BLOCKS, THREADS, 0, stream>>>(
        scores, pos_indices, neg_indices, pos_counts, neg_counts,
        partial_sum, partial_cnt);

    mpcl_reduce_kernel<<<1, 256, 0, stream>>>(
        partial_sum, partial_cnt, (float*)d_out);
}